// TopK_17532056502597
// MI455X (gfx1250) — compile-verified
//
#include <hip/hip_runtime.h>
#include <stdint.h>

typedef __attribute__((ext_vector_type(2))) float v2f;
typedef __attribute__((ext_vector_type(8))) float v8f;

// ---- monotone float <-> uint32 order-preserving mapping ----------------------
static __device__ __forceinline__ uint32_t f2ord(float f) {
  uint32_t b = __float_as_uint(f);
  return b ^ ((uint32_t)(((int32_t)b) >> 31) | 0x80000000u);
}
static __device__ __forceinline__ float ord2f(uint32_t u) {
  uint32_t b = (u & 0x80000000u) ? (u & 0x7FFFFFFFu) : ~u;
  return __uint_as_float(b);
}

// ---- init: ||scorer||, zero state/hist/counter -------------------------------
__global__ void tk_init(const float* __restrict__ scorer, int feats,
                        float* __restrict__ inv_norm, uint32_t* __restrict__ state,
                        uint32_t* __restrict__ hist, uint32_t* __restrict__ counter, int k) {
  __shared__ float red[256];
  float acc = 0.f;
  for (int t = threadIdx.x; t < feats; t += 256) { float x = scorer[t]; acc += x * x; }
  red[threadIdx.x] = acc;
  __syncthreads();
  for (int s = 128; s > 0; s >>= 1) {
    if ((int)threadIdx.x < s) red[threadIdx.x] += red[threadIdx.x + s];
    __syncthreads();
  }
  if (threadIdx.x == 0) {
    *inv_norm = 1.0f / sqrtf(red[0]);
    state[0] = 0u;            // radix-select prefix
    state[1] = (uint32_t)k;   // remaining
    state[2] = 0u;            // threshold
    *counter = 0u;
  }
  if (threadIdx.x < 256) hist[threadIdx.x] = 0u;
}

// ---- scores via V_WMMA_F32_16X16X4_F32 ---------------------------------------
// A (16x4): every row = scorer chunk  -> lane needs scorer[kc + (lane>>4)*2 .. +1]
// B (4x16): column n = emb row (g*16+n) chunk -> lane loads its row's float2
// D: all 16 rows identical -> c[0] on each lane = score of row g*16 + (lane&15)
__global__ void tk_score_wmma(const float* __restrict__ emb, const float* __restrict__ scorer,
                              uint32_t* __restrict__ ukeys, const float* __restrict__ inv_norm_p,
                              int feats, int ngroups) {
  __shared__ float sc[256];
  for (int t = threadIdx.x; t < feats; t += blockDim.x) sc[t] = scorer[t];
  __syncthreads();
  const float inv_norm = *inv_norm_p;
  const int lane = threadIdx.x & 31;
  const int wave = threadIdx.x >> 5;
  const int wavesTotal = gridDim.x * (blockDim.x >> 5);
  const int m = lane & 15;
  const int koff = (lane >> 4) << 1;  // 0 or 2
  for (int g = blockIdx.x * (blockDim.x >> 5) + wave; g < ngroups; g += wavesTotal) {
    const int row = g * 16 + m;
    const float* rp = emb + (size_t)row * feats;
    v8f c = {0.f, 0.f, 0.f, 0.f, 0.f, 0.f, 0.f, 0.f};
    for (int kc = 0; kc < feats; kc += 4) {
      v2f a = *(const v2f*)(sc + kc + koff);   // broadcast within half-wave
      v2f b = *(const v2f*)(rp + kc + koff);   // global_load_b64, fully coalesced
      c = __builtin_amdgcn_wmma_f32_16x16x4_f32(false, a, false, b, (short)0, c,
                                                false, false);
    }
    if (lane < 16) ukeys[row] = f2ord(c[0] * inv_norm);
  }
}

// scalar tail for nrows % 16 (not hit for N=2M)
__global__ void tk_score_tail(const float* __restrict__ emb, const float* __restrict__ scorer,
                              uint32_t* __restrict__ ukeys, const float* __restrict__ inv_norm_p,
                              int feats, int base, int count) {
  int r = threadIdx.x;
  if (r >= count) return;
  const float* rp = emb + (size_t)(base + r) * feats;
  float acc = 0.f;
  for (int j = 0; j < feats; ++j) acc += rp[j] * scorer[j];
  ukeys[base + r] = f2ord(acc * (*inv_norm_p));
}

// ---- MSB-first radix select: 4 rounds of 8-bit histograms --------------------
__global__ void tk_hist(const uint32_t* __restrict__ ukeys, uint32_t* __restrict__ hist,
                        const uint32_t* __restrict__ state, int round, int n) {
  __shared__ uint32_t h[256];
  if (threadIdx.x < 256) h[threadIdx.x] = 0u;
  __syncthreads();
  const uint32_t prefix = state[0];
  const int shift = 24 - 8 * round;
  const uint32_t maskhi = (round == 0) ? 0u : (0xFFFFFFFFu << (shift + 8));
  for (int i = blockIdx.x * blockDim.x + threadIdx.x; i < n; i += gridDim.x * blockDim.x) {
    const uint32_t u = ukeys[i];
    if ((u & maskhi) == prefix) atomicAdd(&h[(u >> shift) & 255u], 1u);
  }
  __syncthreads();
  if (threadIdx.x < 256 && h[threadIdx.x]) atomicAdd(&hist[threadIdx.x], h[threadIdx.x]);
}

__global__ void tk_select(uint32_t* __restrict__ hist, uint32_t* __restrict__ state, int round) {
  if (threadIdx.x == 0) {
    uint32_t remaining = state[1];
    const int shift = 24 - 8 * round;
    uint64_t acc = 0;
    int chosen = 0;
    for (int b = 255; b >= 0; --b) {
      uint32_t cnt = hist[b];
      if (acc + cnt >= (uint64_t)remaining) { chosen = b; break; }
      acc += cnt;
    }
    state[0] |= ((uint32_t)chosen) << shift;
    state[1] = remaining - (uint32_t)acc;
    if (round == 3) state[2] = state[0];  // exact 32-bit threshold
  }
  __syncthreads();
  if (threadIdx.x < 256) hist[threadIdx.x] = 0u;  // clean for next round / replay
}

// ---- compact survivors (ukey >= T) into composite keys ----------------------
// asc composite = (~ukey)<<32 | idx  : ascending <=> score desc, ties idx asc
__global__ void tk_compact(const uint32_t* __restrict__ ukeys, const uint32_t* __restrict__ state,
                           uint64_t* __restrict__ sel, uint32_t* __restrict__ counter,
                           int n, int cap) {
  const uint32_t T = state[2];
  for (int i = blockIdx.x * blockDim.x + threadIdx.x; i < n; i += gridDim.x * blockDim.x) {
    const uint32_t u = ukeys[i];
    if (u >= T) {
      uint32_t pos = atomicAdd(counter, 1u);
      if ((int)pos < cap) sel[pos] = ((uint64_t)(~u) << 32) | (uint32_t)i;
    }
  }
}

// ---- exact rank by counting (keys unique via idx) ----------------------------
__global__ void tk_rank(const uint64_t* __restrict__ sel, const uint32_t* __restrict__ counter,
                        int cap, int k, int* __restrict__ order_idx,
                        float* __restrict__ order_gate) {
  const int M = min((int)*counter, cap);
  if ((int)(blockIdx.x * blockDim.x) >= M) return;  // uniform per block
  __shared__ uint64_t tile[1024];
  const int j = blockIdx.x * blockDim.x + threadIdx.x;
  const bool have = j < M;
  const uint64_t my = have ? sel[j] : ~0ull;
  int rank = 0;
  for (int base = 0; base < M; base += 1024) {
    const int cnt = min(1024, M - base);
    __syncthreads();
    for (int t = threadIdx.x; t < cnt; t += blockDim.x) tile[t] = sel[base + t];
    __syncthreads();
    int t = 0;
    for (; t + 4 <= cnt; t += 4) {
      rank += (int)(tile[t] < my) + (int)(tile[t + 1] < my) +
              (int)(tile[t + 2] < my) + (int)(tile[t + 3] < my);
    }
    for (; t < cnt; ++t) rank += (int)(tile[t] < my);
  }
  if (have && rank < k) {
    const uint32_t idx = (uint32_t)my;
    const uint32_t u = ~(uint32_t)(my >> 32);
    order_idx[rank] = (int)idx;
    order_gate[rank] = tanhf(ord2f(u));
  }
}

// ---- gather + gate + transpose: 32 columns per block via padded LDS ----------
__global__ void tk_gather(const float* __restrict__ emb, const int* __restrict__ order_idx,
                          const float* __restrict__ order_gate, float* __restrict__ out,
                          int feats, int k) {
  __shared__ float tile[32 * 160];
  __shared__ int idxs[32];
  __shared__ float gates[32];
  const int stride = feats + 1;  // 137 -> gcd(137,64)=1, conflict-free transpose
  const int col0 = blockIdx.x * 32;
  const int ncol = min(32, k - col0);
  if ((int)threadIdx.x < 32) {
    const int c = col0 + (int)threadIdx.x;
    const bool v = (int)threadIdx.x < ncol;
    idxs[threadIdx.x] = v ? order_idx[c] : 0;
    gates[threadIdx.x] = v ? order_gate[c] : 0.f;
  }
  __syncthreads();
  const int total = 32 * feats;
  for (int e = threadIdx.x; e < total; e += blockDim.x) {
    const int r = e / feats, f = e - r * feats;
    tile[r * stride + f] = emb[(size_t)idxs[r] * feats + f];  // coalesced row reads
  }
  __syncthreads();
  const int lane = threadIdx.x & 31;
  const int nw = blockDim.x >> 5;
  if (lane < ncol) {
    const float g = gates[lane];
    for (int f = threadIdx.x >> 5; f < feats; f += nw)
      out[(size_t)f * k + col0 + lane] = tile[lane * stride + f] * g;  // coalesced writes
  }
}

// ---- launcher ----------------------------------------------------------------
extern "C" void kernel_launch(void* const* d_in, const int* in_sizes, int n_in,
                              void* d_out, int out_size, void* d_ws, size_t ws_size,
                              hipStream_t stream) {
  const float* emb = (const float*)d_in[0];
  const float* scorer = (const float*)d_in[1];
  float* out = (float*)d_out;
  const int feats = in_sizes[1];          // 136
  const int n = in_sizes[0] / feats;      // 2,000,000
  const int k = out_size / feats;         // 50,000 (device scalar unreadable in capture)
  const int cap = k + 16384;              // slack for threshold ties

  uintptr_t p = (uintptr_t)d_ws;
  auto alloc = [&](size_t bytes) -> void* {
    uintptr_t r = p;
    p += (bytes + 255) & ~(size_t)255;
    return (void*)r;
  };
  uint32_t* ukeys = (uint32_t*)alloc((size_t)n * 4);
  uint64_t* sel = (uint64_t*)alloc((size_t)cap * 8);
  int* order_idx = (int*)alloc((size_t)k * 4);
  float* order_gate = (float*)alloc((size_t)k * 4);
  uint32_t* hist = (uint32_t*)alloc(256 * 4);
  uint32_t* state = (uint32_t*)alloc(64);
  uint32_t* counter = state + 8;
  float* inv_norm = (float*)(state + 12);

  tk_init<<<1, 256, 0, stream>>>(scorer, feats, inv_norm, state, hist, counter, k);

  const int ngroups = n / 16;
  const int tail = n - ngroups * 16;
  int sblocks = (ngroups + 7) / 8;
  if (sblocks > 2048) sblocks = 2048;
  if (sblocks > 0)
    tk_score_wmma<<<sblocks, 256, 0, stream>>>(emb, scorer, ukeys, inv_norm, feats, ngroups);
  if (tail > 0)
    tk_score_tail<<<1, 32, 0, stream>>>(emb, scorer, ukeys, inv_norm, feats, ngroups * 16, tail);

  for (int r = 0; r < 4; ++r) {
    tk_hist<<<1024, 256, 0, stream>>>(ukeys, hist, state, r, n);
    tk_select<<<1, 256, 0, stream>>>(hist, state, r);
  }
  tk_compact<<<1024, 256, 0, stream>>>(ukeys, state, sel, counter, n, cap);
  tk_rank<<<(cap + 255) / 256, 256, 0, stream>>>(sel, counter, cap, k, order_idx, order_gate);
  tk_gather<<<(k + 31) / 32, 256, 0, stream>>>(emb, order_idx, order_gate, out, feats, k);
}